// DecoderCrossAttention_18674517803216
// MI455X (gfx1250) — compile-verified
//
#include <hip/hip_runtime.h>
#include <hip/hip_bf16.h>

typedef float v2f __attribute__((ext_vector_type(2)));
typedef float v8f __attribute__((ext_vector_type(8)));

// ---------------------------------------------------------------------------
// LayerNorm: one 256-thread block per row of 1024 elements.
// ---------------------------------------------------------------------------
__global__ __launch_bounds__(256)
void layernorm_kernel(const float* __restrict__ x,
                      const float* __restrict__ gamma,
                      const float* __restrict__ beta,
                      float* __restrict__ y) {
  constexpr int E = 1024;
  const long row = blockIdx.x;
  const float* xr = x + row * E;
  float* yr = y + row * E;
  const int tid = threadIdx.x;

  float v[4];
  float s = 0.f, ss = 0.f;
#pragma unroll
  for (int i = 0; i < 4; ++i) {
    v[i] = xr[tid + i * 256];
    s += v[i];
    ss += v[i] * v[i];
  }
#pragma unroll
  for (int off = 16; off > 0; off >>= 1) {
    s  += __shfl_xor(s, off, 32);
    ss += __shfl_xor(ss, off, 32);
  }
  __shared__ float sh_s[8], sh_ss[8];
  const int w = tid >> 5, ln = tid & 31;
  if (ln == 0) { sh_s[w] = s; sh_ss[w] = ss; }
  __syncthreads();
  if (tid == 0) {
    float a = 0.f, b = 0.f;
#pragma unroll
    for (int i = 0; i < 8; ++i) { a += sh_s[i]; b += sh_ss[i]; }
    sh_s[0] = a; sh_ss[0] = b;
  }
  __syncthreads();
  const float mu = sh_s[0] * (1.0f / E);
  const float var = sh_ss[0] * (1.0f / E) - mu * mu;
  const float rstd = rsqrtf(var + 1e-5f);
#pragma unroll
  for (int i = 0; i < 4; ++i) {
    const int j = tid + i * 256;
    yr[j] = (v[i] - mu) * rstd * gamma[j] + beta[j];
  }
}

// ---------------------------------------------------------------------------
// Softmax with scale (1/sqrt(Dh)) and non-standard mask fill (1e-10), in place
// on the attn output region. One 256-thread block per row of 1024.
// ---------------------------------------------------------------------------
__global__ __launch_bounds__(256)
void softmax_mask_kernel(float* __restrict__ attn,
                         const int* __restrict__ klen,
                         int rowsPerBatch) {
  constexpr int Skv = 1024;
  const long row = blockIdx.x;
  const int b = (int)(row / rowsPerBatch);
  float* p = attn + row * Skv;
  const int kl = klen[b];
  const int tid = threadIdx.x;

  float lv[4];
  float mx = -3.4e38f;
#pragma unroll
  for (int i = 0; i < 4; ++i) {
    const int j = tid + i * 256;
    const float raw = p[j];
    const float val = (j < kl) ? raw * 0.125f : 1e-10f;
    lv[i] = val;
    mx = fmaxf(mx, val);
  }
#pragma unroll
  for (int off = 16; off > 0; off >>= 1) mx = fmaxf(mx, __shfl_xor(mx, off, 32));
  __shared__ float sh[8];
  const int w = tid >> 5, ln = tid & 31;
  if (ln == 0) sh[w] = mx;
  __syncthreads();
  if (tid == 0) {
    float m = sh[0];
#pragma unroll
    for (int i = 1; i < 8; ++i) m = fmaxf(m, sh[i]);
    sh[0] = m;
  }
  __syncthreads();
  mx = sh[0];
  __syncthreads();

  float sum = 0.f;
#pragma unroll
  for (int i = 0; i < 4; ++i) {
    lv[i] = expf(lv[i] - mx);
    sum += lv[i];
  }
#pragma unroll
  for (int off = 16; off > 0; off >>= 1) sum += __shfl_xor(sum, off, 32);
  if (ln == 0) sh[w] = sum;
  __syncthreads();
  if (tid == 0) {
    float a = 0.f;
#pragma unroll
    for (int i = 0; i < 8; ++i) a += sh[i];
    sh[0] = a;
  }
  __syncthreads();
  const float inv = 1.0f / sh[0];
#pragma unroll
  for (int i = 0; i < 4; ++i) p[tid + i * 256] = lv[i] * inv;
}

// ---------------------------------------------------------------------------
// Tiled fp32 WMMA GEMM with double-buffered LDS and K-contiguous tile layout
// (every fragment is one aligned ds_load_b64, bank-conflict free).
//   C = A * B  (TRANSB: B given as (N x K) row-major)
// Batched-head pointer offsets: ptr += (z/H)*stride_b + (z%H)*stride_h.
// EPI==1: C = (acc + bias[n]) * (row_in_batch < rowLen[batch]) + residual.
// Assumes M%BM==0, N%BN==0, K%BK==0.
// ---------------------------------------------------------------------------
template <int BM, int BN, int BK, int WM, int WN, int TRANSB, int EPI>
__global__ __launch_bounds__(32 * (BM / WM) * (BN / WN))
void gemm_f32_wmma(const float* __restrict__ A, int lda, long sAb, long sAh,
                   const float* __restrict__ B, int ldb, long sBb, long sBh,
                   float* __restrict__ C, int ldc, long sCb, long sCh,
                   int K, int H,
                   const float* __restrict__ bias,
                   const float* __restrict__ residual,
                   const int* __restrict__ rowLen, int rowsPerBatch) {
  constexpr int THREADS = 32 * (BM / WM) * (BN / WN);
  constexpr int MI = WM / 16;
  constexpr int NI = WN / 16;
  constexpr int NWN = BN / WN;
  constexpr int KP = BK + 4;  // 36 floats: 16B-aligned rows, conflict-free b64
  constexpr int AITER = (BM * BK / 4) / THREADS;
  constexpr int BITER = (BN * BK / 4) / THREADS;

  const int z = blockIdx.z;
  const int zb = z / H, zh = z % H;
  A += zb * sAb + zh * sAh;
  B += zb * sBb + zh * sBh;
  C += zb * sCb + zh * sCh;

  const int mBlock = blockIdx.y * BM;
  const int nBlock = blockIdx.x * BN;

  __shared__ alignas(16) float As[2][BM][KP];
  __shared__ alignas(16) float Bs[2][BN][KP];

  const int tid = threadIdx.x;
  const int wid = tid >> 5;
  const int lane = tid & 31;
  const int half = lane >> 4;  // 0: K pair {0,1}; 1: K pair {2,3}
  const int lid = lane & 15;
  const int wm = (wid / NWN) * WM;
  const int wn = (wid % NWN) * WN;

  v8f acc[MI][NI] = {};
  float4 aReg[AITER];
  float4 bReg[BITER];

  // ---- global -> registers for one K-tile ----
  auto globalLoad = [&](int kk) {
#pragma unroll
    for (int it = 0; it < AITER; ++it) {
      const int q = tid + it * THREADS;
      const int m = q / (BK / 4);
      const int kq = (q % (BK / 4)) * 4;
      aReg[it] = *(const float4*)(A + (long)(mBlock + m) * lda + kk + kq);
    }
#pragma unroll
    for (int it = 0; it < BITER; ++it) {
      const int q = tid + it * THREADS;
      if (TRANSB) {
        const int n = q / (BK / 4);
        const int kq = (q % (BK / 4)) * 4;
        bReg[it] = *(const float4*)(B + (long)(nBlock + n) * ldb + kk + kq);
      } else {
        const int k = q / (BN / 4);
        const int nq = (q % (BN / 4)) * 4;
        bReg[it] = *(const float4*)(B + (long)(kk + k) * ldb + nBlock + nq);
      }
    }
  };

  // ---- registers -> LDS (K-contiguous layout) ----
  auto ldsStore = [&](int buf) {
#pragma unroll
    for (int it = 0; it < AITER; ++it) {
      const int q = tid + it * THREADS;
      const int m = q / (BK / 4);
      const int kq = (q % (BK / 4)) * 4;
      *(float4*)&As[buf][m][kq] = aReg[it];  // direct copy, no transpose
    }
#pragma unroll
    for (int it = 0; it < BITER; ++it) {
      const int q = tid + it * THREADS;
      if (TRANSB) {
        const int n = q / (BK / 4);
        const int kq = (q % (BK / 4)) * 4;
        *(float4*)&Bs[buf][n][kq] = bReg[it];  // direct copy
      } else {
        const int k = q / (BN / 4);
        const int nq = (q % (BN / 4)) * 4;
        Bs[buf][nq + 0][k] = bReg[it].x;  // transpose scatter
        Bs[buf][nq + 1][k] = bReg[it].y;
        Bs[buf][nq + 2][k] = bReg[it].z;
        Bs[buf][nq + 3][k] = bReg[it].w;
      }
    }
  };

  // ---- WMMA over one K-tile: fragments are contiguous ds_load_b64 ----
  auto compute = [&](int buf) {
#pragma unroll
    for (int k0 = 0; k0 < BK; k0 += 4) {
      v2f af[MI];
      v2f bf[NI];
#pragma unroll
      for (int mi = 0; mi < MI; ++mi)
        af[mi] = *(const v2f*)&As[buf][wm + mi * 16 + lid][k0 + 2 * half];
#pragma unroll
      for (int ni = 0; ni < NI; ++ni)
        bf[ni] = *(const v2f*)&Bs[buf][wn + ni * 16 + lid][k0 + 2 * half];
#pragma unroll
      for (int mi = 0; mi < MI; ++mi)
#pragma unroll
        for (int ni = 0; ni < NI; ++ni)
          acc[mi][ni] = __builtin_amdgcn_wmma_f32_16x16x4_f32(
              false, af[mi], false, bf[ni], (short)0, acc[mi][ni], false, false);
    }
  };

  // ---- pipelined main loop: 1 barrier per K-tile ----
  globalLoad(0);
  ldsStore(0);
  __syncthreads();

  const int ktiles = K / BK;
  for (int t = 0; t < ktiles; ++t) {
    const int cur = t & 1;
    if (t + 1 < ktiles) {
      globalLoad((t + 1) * BK);
      if (t + 2 < ktiles) {  // deep prefetch of tile t+2 toward L2
        __builtin_prefetch(
            A + (long)(mBlock + (tid & (BM - 1))) * lda + (t + 2) * BK, 0, 1);
        if (TRANSB)
          __builtin_prefetch(
              B + (long)(nBlock + (tid & (BN - 1))) * ldb + (t + 2) * BK, 0, 1);
        else
          __builtin_prefetch(
              B + (long)((t + 2) * BK + (tid & (BK - 1))) * ldb + nBlock, 0, 1);
      }
    }
    compute(cur);
    if (t + 1 < ktiles) {
      ldsStore(cur ^ 1);
      __syncthreads();
    }
  }

  // ---- epilogue ----
#pragma unroll
  for (int mi = 0; mi < MI; ++mi) {
#pragma unroll
    for (int ni = 0; ni < NI; ++ni) {
      const int n = nBlock + wn + ni * 16 + lid;
#pragma unroll
      for (int i = 0; i < 8; ++i) {
        const int m = mBlock + wm + mi * 16 + half * 8 + i;
        float v = acc[mi][ni][i];
        if (EPI == 1) {
          v += bias[n];
          const int bidx = m / rowsPerBatch;
          const int r = m - bidx * rowsPerBatch;
          v = (r < rowLen[bidx]) ? v : 0.0f;
          v += residual[(long)m * ldc + n];
        }
        C[(long)m * ldc + n] = v;
      }
    }
  }
}

// ---------------------------------------------------------------------------
// Launch
// ---------------------------------------------------------------------------
extern "C" void kernel_launch(void* const* d_in, const int* in_sizes, int n_in,
                              void* d_out, int out_size, void* d_ws, size_t ws_size,
                              hipStream_t stream) {
  const float* query = (const float*)d_in[0];
  const float* key   = (const float*)d_in[1];
  const float* value = (const float*)d_in[2];
  const int*   qlen  = (const int*)d_in[3];
  const int*   klen  = (const int*)d_in[4];
  const float* w_q   = (const float*)d_in[5];
  const float* w_k   = (const float*)d_in[6];
  const float* w_v   = (const float*)d_in[7];
  const float* gamma = (const float*)d_in[8];
  const float* beta  = (const float*)d_in[9];
  const float* out_w = (const float*)d_in[10];
  const float* out_b = (const float*)d_in[11];

  constexpr int B = 4, Sq = 1024, Skv = 1024, E = 1024, D = 1024, H = 16, Dh = 64;
  constexpr long rows = (long)B * Sq;   // 4096
  constexpr long mat = rows * D;        // 4Mi elements, 16 MB each

  float* ws = (float*)d_ws;
  float* qn = ws;            // normalized query (also residual)
  float* Qm = qn + mat;      // q projection
  float* Km = Qm + mat;      // k projection
  float* Vm = Km + mat;      // v projection
  float* Cx = Vm + mat;      // merged context

  float* out_skip = (float*)d_out;        // B*Sq*E
  float* attn = out_skip + mat;           // B*H*Sq*Skv

  // 1) LayerNorm(query) -> qn
  layernorm_kernel<<<dim3((unsigned)rows), 256, 0, stream>>>(query, gamma, beta, qn);

  // 2) Projections: (B*S, E) @ (E, D)
  gemm_f32_wmma<128, 128, 32, 64, 32, 0, 0>
      <<<dim3(D / 128, (unsigned)(rows / 128), 1), 256, 0, stream>>>(
          qn, E, 0, 0, w_q, D, 0, 0, Qm, D, 0, 0, E, 1, nullptr, nullptr, nullptr, 1);
  gemm_f32_wmma<128, 128, 32, 64, 32, 0, 0>
      <<<dim3(D / 128, (unsigned)(rows / 128), 1), 256, 0, stream>>>(
          key, E, 0, 0, w_k, D, 0, 0, Km, D, 0, 0, E, 1, nullptr, nullptr, nullptr, 1);
  gemm_f32_wmma<128, 128, 32, 64, 32, 0, 0>
      <<<dim3(D / 128, (unsigned)(rows / 128), 1), 256, 0, stream>>>(
          value, E, 0, 0, w_v, D, 0, 0, Vm, D, 0, 0, E, 1, nullptr, nullptr, nullptr, 1);

  // 3) Scores per (b,h): q_h (Sq x Dh) @ k_h^T -> raw scores into attn region.
  //    Head matrices are contiguous (the reference's view-reshape): stride Sq*Dh.
  gemm_f32_wmma<128, 128, 32, 64, 32, 1, 0>
      <<<dim3(Skv / 128, Sq / 128, B * H), 256, 0, stream>>>(
          Qm, Dh, (long)H * Sq * Dh, (long)Sq * Dh,
          Km, Dh, (long)H * Skv * Dh, (long)Skv * Dh,
          attn, Skv, (long)H * Sq * Skv, (long)Sq * Skv,
          Dh, H, nullptr, nullptr, nullptr, 1);

  // 4) scale + mask-fill(1e-10) + softmax, in place
  softmax_mask_kernel<<<dim3(B * H * Sq), 256, 0, stream>>>(attn, klen, H * Sq);

  // 5) ctx per (b,h): attn (Sq x Skv) @ v_h (Skv x Dh) -> columns h*64.. of Cx
  gemm_f32_wmma<128, 64, 32, 64, 16, 0, 0>
      <<<dim3(1, Sq / 128, B * H), 256, 0, stream>>>(
          attn, Skv, (long)H * Sq * Skv, (long)Sq * Skv,
          Vm, Dh, (long)H * Skv * Dh, (long)Skv * Dh,
          Cx, D, (long)Sq * D, (long)Dh,
          Skv, H, nullptr, nullptr, nullptr, 1);

  // 6) Output projection + bias + query-length mask + residual(qn)
  gemm_f32_wmma<128, 128, 32, 64, 32, 0, 1>
      <<<dim3(E / 128, (unsigned)(rows / 128), 1), 256, 0, stream>>>(
          Cx, D, 0, 0, out_w, E, 0, 0, out_skip, E, 0, 0,
          D, 1, out_b, qn, qlen, Sq);

  (void)in_sizes; (void)n_in; (void)out_size; (void)ws_size;
}